// ArapLoss_34110630265646
// MI455X (gfx1250) — compile-verified
//
#include <hip/hip_runtime.h>
#include <hip/hip_bf16.h>
#include <math.h>

#define BATCH    16
#define BLOCK    256
#define NWAVES   (BLOCK / 32)
#define MAXD_CAP 8

typedef __attribute__((ext_vector_type(2))) float v2f;
typedef __attribute__((ext_vector_type(8))) float v8f;

__device__ __forceinline__ void cross3(const float a[3], const float b[3], float c[3]) {
    c[0] = a[1]*b[2] - a[2]*b[1];
    c[1] = a[2]*b[0] - a[0]*b[2];
    c[2] = a[0]*b[1] - a[1]*b[0];
}
__device__ __forceinline__ float dot3(const float a[3], const float b[3]) {
    return a[0]*b[0] + a[1]*b[1] + a[2]*b[2];
}

// 32-lane sum via V_WMMA_F32_16X16X4_F32. Lane partials form the 16x4 f32
// A-matrix operand layout (lane M holds K=0/1 in the two A VGPRs, lane 16+M
// holds K=2/3); B = ones(4x16) makes each D row the A row-sum. Summing the 8
// D VGPRs gives sum(acc_0..7 + acc_16..23) in lanes 0-15 and the complement
// in lanes 16-31; one bpermute combines the halves. Valid in lane 0.
// Requires EXEC all-ones (caller must invoke from all 32 lanes).
__device__ __forceinline__ float wave32_sum_wmma(float acc) {
    v2f a;  a[0] = acc;  a[1] = 0.0f;
    v2f bm; bm[0] = 1.0f; bm[1] = 1.0f;
    v8f c = {};
    v8f d = __builtin_amdgcn_wmma_f32_16x16x4_f32(
        /*neg_a=*/false, a, /*neg_b=*/false, bm,
        /*c_mod=*/(short)0, c, /*reuse_a=*/false, /*reuse_b=*/false);
    float rowacc = d[0] + d[1] + d[2] + d[3] + d[4] + d[5] + d[6] + d[7];
    return rowacc + __shfl(rowacc, 16, 32);
}

// Null-space direction of (S - lam*I) via the largest row-pair cross product.
// Returns squared norm of the (unnormalized) vector.
__device__ __forceinline__ float nullvec3(float s00, float s01, float s02,
                                          float s11, float s12, float s22,
                                          float lam, float v[3]) {
    float r0[3] = { s00 - lam, s01, s02 };
    float r1[3] = { s01, s11 - lam, s12 };
    float r2[3] = { s02, s12, s22 - lam };
    float c01[3], c02[3], c12[3];
    cross3(r0, r1, c01);
    cross3(r0, r2, c02);
    cross3(r1, r2, c12);
    float n01 = dot3(c01, c01), n02 = dot3(c02, c02), n12 = dot3(c12, c12);
    float nb = n01;
    v[0] = c01[0]; v[1] = c01[1]; v[2] = c01[2];
    if (n02 > nb) { nb = n02; v[0] = c02[0]; v[1] = c02[1]; v[2] = c02[2]; }
    if (n12 > nb) { nb = n12; v[0] = c12[0]; v[1] = c12[1]; v[2] = c12[2]; }
    return nb;
}

// ---------------------------------------------------------------------------
// Kernel 1: per-(batch, vertex) ARAP energy; per-wave WMMA reduction, then a
// single tiny LDS combine of the 8 wave sums.
//   pred : (B, 3, n) f32      adj_idx : (n, maxd) i32   adj_w : (n, maxd) f32
//   tevT : (n, 3, maxd) f32   tevw    : (n, maxd, 3) f32
//   partials : (B, gridDim.x) f32
// ---------------------------------------------------------------------------
__global__ __launch_bounds__(BLOCK)
void arap_vertex_kernel(const float* __restrict__ pred,
                        const int*   __restrict__ adj_idx,
                        const float* __restrict__ adj_w,
                        const float* __restrict__ tevT,
                        const float* __restrict__ tevw,
                        float*       __restrict__ partials,
                        int n, int maxd, float nrg_clamp)
{
    const int b = blockIdx.y;
    const int v = blockIdx.x * BLOCK + threadIdx.x;

    float nrg = 0.0f;
    if (v < n) {
        const float* pb = pred + (size_t)b * 3u * (size_t)n;
        const float p0x = pb[v];
        const float p0y = pb[(size_t)n + v];
        const float p0z = pb[2u * (size_t)n + v];

        float pe[3][MAXD_CAP];
        float A[3][3] = { {0,0,0}, {0,0,0}, {0,0,0} };

        #pragma unroll
        for (int j = 0; j < MAXD_CAP; ++j) {
            if (j < maxd) {
                const int idx = adj_idx[(size_t)v * maxd + j];
                const float ex = pb[idx]                  - p0x;
                const float ey = pb[(size_t)n + idx]      - p0y;
                const float ez = pb[2u * (size_t)n + idx] - p0z;
                pe[0][j] = ex; pe[1][j] = ey; pe[2][j] = ez;
                const float* tw = tevw + ((size_t)v * maxd + j) * 3u;
                const float t0 = tw[0], t1 = tw[1], t2 = tw[2];
                A[0][0] = fmaf(ex, t0, A[0][0]); A[0][1] = fmaf(ex, t1, A[0][1]); A[0][2] = fmaf(ex, t2, A[0][2]);
                A[1][0] = fmaf(ey, t0, A[1][0]); A[1][1] = fmaf(ey, t1, A[1][1]); A[1][2] = fmaf(ey, t2, A[1][2]);
                A[2][0] = fmaf(ez, t0, A[2][0]); A[2][1] = fmaf(ez, t1, A[2][1]); A[2][2] = fmaf(ez, t2, A[2][2]);
            }
        }

        const float stab = 1000.0f;
        #pragma unroll
        for (int i = 0; i < 3; ++i)
            #pragma unroll
            for (int j = 0; j < 3; ++j)
                A[i][j] *= stab;

        // S = A^T A (exactly symmetric by construction)
        float s00 = A[0][0]*A[0][0] + A[1][0]*A[1][0] + A[2][0]*A[2][0];
        float s01 = A[0][0]*A[0][1] + A[1][0]*A[1][1] + A[2][0]*A[2][1];
        float s02 = A[0][0]*A[0][2] + A[1][0]*A[1][2] + A[2][0]*A[2][2];
        float s11 = A[0][1]*A[0][1] + A[1][1]*A[1][1] + A[2][1]*A[2][1];
        float s12 = A[0][1]*A[0][2] + A[1][1]*A[1][2] + A[2][1]*A[2][2];
        float s22 = A[0][2]*A[0][2] + A[1][2]*A[1][2] + A[2][2]*A[2][2];

        // Normalize magnitude for numerically stable eigensolve
        float f = fmaxf(fmaxf(fmaxf(fabsf(s00), fabsf(s11)), fabsf(s22)),
                        fmaxf(fmaxf(fabsf(s01), fabsf(s02)), fabsf(s12)));
        float lam0, lam1, lam2;            // ascending, in normalized units
        float Vc[3][3];                    // columns = eigenvectors v0|v1|v2
        if (f < 1e-30f) {
            lam0 = lam1 = lam2 = 0.0f;
            Vc[0][0]=1; Vc[1][0]=0; Vc[2][0]=0;
            Vc[0][1]=0; Vc[1][1]=1; Vc[2][1]=0;
            Vc[0][2]=0; Vc[1][2]=0; Vc[2][2]=1;
            f = 1.0f;
        } else {
            const float inv_f = 1.0f / f;
            const float a00 = s00*inv_f, a01 = s01*inv_f, a02 = s02*inv_f;
            const float a11 = s11*inv_f, a12 = s12*inv_f, a22 = s22*inv_f;

            const float p1 = a01*a01 + a02*a02 + a12*a12;
            const float q  = (a00 + a11 + a22) * (1.0f/3.0f);
            const float p2 = (a00-q)*(a00-q) + (a11-q)*(a11-q) + (a22-q)*(a22-q) + 2.0f*p1;
            const float p  = sqrtf(fmaxf(p2 * (1.0f/6.0f), 0.0f));

            if (p < 1e-12f) {
                lam0 = lam1 = lam2 = q;    // isotropic
            } else {
                const float ip  = 1.0f / p;
                const float b00 = (a00-q)*ip, b11 = (a11-q)*ip, b22 = (a22-q)*ip;
                const float b01 = a01*ip,     b02 = a02*ip,     b12 = a12*ip;
                float r = 0.5f * ( b00*(b11*b22 - b12*b12)
                                 - b01*(b01*b22 - b12*b02)
                                 + b02*(b01*b12 - b11*b02) );
                r = fminf(fmaxf(r, -1.0f), 1.0f);
                const float phi = acosf(r) * (1.0f/3.0f);
                lam2 = q + 2.0f*p*cosf(phi);                       // largest
                lam0 = q + 2.0f*p*cosf(phi + 2.0943951023931953f); // smallest
                lam1 = 3.0f*q - lam2 - lam0;                       // middle
            }

            // v2: largest-eigenvalue direction
            float v2v[3];
            float nb2 = nullvec3(a00, a01, a02, a11, a12, a22, lam2, v2v);
            if (nb2 < 1e-24f) { v2v[0] = 1.0f; v2v[1] = 0.0f; v2v[2] = 0.0f; }
            {
                const float inv = rsqrtf(fmaxf(dot3(v2v, v2v), 1e-30f));
                v2v[0] *= inv; v2v[1] *= inv; v2v[2] *= inv;
            }
            // v0: smallest, orthogonalized against v2
            float v0v[3];
            nullvec3(a00, a01, a02, a11, a12, a22, lam0, v0v);
            const float d02 = dot3(v0v, v2v);
            v0v[0] -= d02 * v2v[0]; v0v[1] -= d02 * v2v[1]; v0v[2] -= d02 * v2v[2];
            float n0 = dot3(v0v, v0v);
            if (n0 < 1e-24f) {
                float ax[3] = { 1.0f, 0.0f, 0.0f };
                if (fabsf(v2v[0]) > 0.9f) { ax[0] = 0.0f; ax[1] = 1.0f; }
                cross3(v2v, ax, v0v);
                n0 = dot3(v0v, v0v);
            }
            {
                const float inv = rsqrtf(fmaxf(n0, 1e-30f));
                v0v[0] *= inv; v0v[1] *= inv; v0v[2] *= inv;
            }
            float v1v[3];
            cross3(v2v, v0v, v1v);   // unit, orthogonal -> middle eigenvector

            Vc[0][0]=v0v[0]; Vc[1][0]=v0v[1]; Vc[2][0]=v0v[2];
            Vc[0][1]=v1v[0]; Vc[1][1]=v1v[1]; Vc[2][1]=v1v[2];
            Vc[0][2]=v2v[0]; Vc[1][2]=v2v[1]; Vc[2][2]=v2v[2];
        }

        // singular values, clipped as in reference: sqrt(clip(lam, 1e-12*stab^2))
        float sv[3];
        sv[0] = sqrtf(fmaxf(lam0 * f, 1e-6f));
        sv[1] = sqrtf(fmaxf(lam1 * f, 1e-6f));
        sv[2] = sqrtf(fmaxf(lam2 * f, 1e-6f));

        // U = A V / s  (column-wise)
        float U[3][3];
        #pragma unroll
        for (int k = 0; k < 3; ++k) {
            const float is = 1.0f / sv[k];
            #pragma unroll
            for (int i = 0; i < 3; ++i)
                U[i][k] = (A[i][0]*Vc[0][k] + A[i][1]*Vc[1][k] + A[i][2]*Vc[2][k]) * is;
        }

        // Rt = U V^T ; d = det(Rt)
        float Rt[3][3];
        #pragma unroll
        for (int i = 0; i < 3; ++i)
            #pragma unroll
            for (int j = 0; j < 3; ++j)
                Rt[i][j] = U[i][0]*Vc[j][0] + U[i][1]*Vc[j][1] + U[i][2]*Vc[j][2];
        const float det = Rt[0][0]*(Rt[1][1]*Rt[2][2] - Rt[2][1]*Rt[1][2])
                        - Rt[1][0]*(Rt[0][1]*Rt[2][2] - Rt[2][1]*Rt[0][2])
                        + Rt[2][0]*(Rt[0][1]*Rt[1][2] - Rt[1][1]*Rt[0][2]);

        // R = U diag(1,1,det) V^T
        float R[3][3];
        #pragma unroll
        for (int i = 0; i < 3; ++i)
            #pragma unroll
            for (int j = 0; j < 3; ++j)
                R[i][j] = U[i][0]*Vc[j][0] + U[i][1]*Vc[j][1] + det*U[i][2]*Vc[j][2];

        // nrg = sum_j w_j * || pe_j - R t_j ||
        const float* tv = tevT + (size_t)v * 3u * (size_t)maxd;
        #pragma unroll
        for (int j = 0; j < MAXD_CAP; ++j) {
            if (j < maxd) {
                const float t0 = tv[j];
                const float t1 = tv[(size_t)maxd + j];
                const float t2 = tv[2u * (size_t)maxd + j];
                const float dx = pe[0][j] - (R[0][0]*t0 + R[0][1]*t1 + R[0][2]*t2);
                const float dy = pe[1][j] - (R[1][0]*t0 + R[1][1]*t1 + R[1][2]*t2);
                const float dz = pe[2][j] - (R[2][0]*t0 + R[2][1]*t1 + R[2][2]*t2);
                nrg += adj_w[(size_t)v * maxd + j] * sqrtf(dx*dx + dy*dy + dz*dz);
            }
        }
        nrg = fminf(nrg, nrg_clamp);
    }

    // Per-wave WMMA reduction (all 32 lanes of every wave reach this point:
    // EXEC all-ones as WMMA requires), then one LDS combine of 8 wave sums.
    const float wsum = wave32_sum_wmma(nrg);

    __shared__ float swave[NWAVES];
    const int wave = threadIdx.x >> 5;
    const int lane = threadIdx.x & 31;
    if (lane == 0) swave[wave] = wsum;
    __syncthreads();
    if (threadIdx.x == 0) {
        float s = 0.0f;
        #pragma unroll
        for (int w = 0; w < NWAVES; ++w) s += swave[w];
        partials[(size_t)b * gridDim.x + blockIdx.x] = s;
    }
}

// ---------------------------------------------------------------------------
// Kernel 2: one wave32 per batch. Strided accumulate of block partials, then
// the same WMMA 32-lane reduction; lane 0 writes the batch mean.
// ---------------------------------------------------------------------------
__global__ __launch_bounds__(32)
void arap_reduce_kernel(const float* __restrict__ partials,
                        float*       __restrict__ out,
                        int numBlocks, int n)
{
    const int b    = blockIdx.x;
    const int lane = threadIdx.x;

    float acc = 0.0f;
    for (int i = lane; i < numBlocks; i += 32)
        acc += partials[(size_t)b * numBlocks + i];

    const float total = wave32_sum_wmma(acc);
    if (lane == 0)
        out[b] = total / (float)n;
}

extern "C" void kernel_launch(void* const* d_in, const int* in_sizes, int n_in,
                              void* d_out, int out_size, void* d_ws, size_t ws_size,
                              hipStream_t stream)
{
    const float* pred    = (const float*)d_in[0];   // (B,3,n) f32
    const int*   adj_idx = (const int*)  d_in[1];   // (n,maxd) i32
    const float* adj_w   = (const float*)d_in[2];   // (n,maxd) f32
    const float* tevT    = (const float*)d_in[3];   // (n,3,maxd) f32
    const float* tevw    = (const float*)d_in[4];   // (n,maxd,3) f32
    float*       out     = (float*)d_out;           // (B,) f32

    const int n = in_sizes[0] / (BATCH * 3);
    int maxd    = in_sizes[1] / n;
    if (maxd > MAXD_CAP) maxd = MAXD_CAP;

    const int nBlocks = (n + BLOCK - 1) / BLOCK;
    float* partials = (float*)d_ws;                 // B * nBlocks floats

    dim3 grid(nBlocks, BATCH);
    arap_vertex_kernel<<<grid, BLOCK, 0, stream>>>(pred, adj_idx, adj_w, tevT, tevw,
                                                   partials, n, maxd, 1.0f);
    arap_reduce_kernel<<<BATCH, 32, 0, stream>>>(partials, out, nBlocks, n);
}